// EGCN_layer_1735166787693
// MI455X (gfx1250) — compile-verified
//
#include <hip/hip_runtime.h>

#define B_   8
#define N_   1024
#define F_   32
#define H_   32
#define P_   4
#define OUT_ 64

typedef float v2f __attribute__((ext_vector_type(2)));
typedef float v8f __attribute__((ext_vector_type(8)));

__device__ inline v8f v8f_zero() { v8f z; for (int i = 0; i < 8; ++i) z[i] = 0.f; return z; }

// workspace layout (float offsets)
#define OFS_GL   0             // B*N*H           = 262144
#define OFS_SA   262144        // P*B*N           = 32768
#define OFS_SB   294912        // P*B*N
#define OFS_R    327680        // B*P*N  (stores 1/rowsum)
#define OFS_C    360448        // B*P*N  (colsum, later inverted)
#define OFS_V2   393216        // P*F
#define OFS_V3   393344        // P*F
#define OFS_BP   393472        // P (fused bias)
#define OFS_S    393728        // B*P*N*N packed fp32 normalized matrix
#define RESX_N   (B_*N_*OUT_)  // 524288 floats of res_x at d_out[0]

// ---------------------------------------------------------------- K0: tiny prep
// v2[p,f] = sum_h W2[p,f,h]*w4a[p,h]; v3 likewise; bias[p] = b4 + b2.w4a + b3.w4b
__global__ void k_prep(const float* W2, const float* b2, const float* W3, const float* b3,
                       const float* w4a, const float* w4b, const float* b4, float* ws) {
  int t = threadIdx.x;
  if (t < P_ * F_) {
    int p = t / F_, f = t % F_;
    float s2 = 0.f, s3 = 0.f;
    for (int h = 0; h < H_; ++h) {
      s2 += W2[(p * F_ + f) * H_ + h] * w4a[p * H_ + h];
      s3 += W3[(p * F_ + f) * H_ + h] * w4b[p * H_ + h];
    }
    ws[OFS_V2 + t] = s2;
    ws[OFS_V3 + t] = s3;
  } else if (t < P_ * F_ + P_) {
    int p = t - P_ * F_;
    float s = b4[p];
    for (int h = 0; h < H_; ++h)
      s += b2[p * H_ + h] * w4a[p * H_ + h] + b3[p * H_ + h] * w4b[p * H_ + h];
    ws[OFS_BP + p] = s;
  }
}

__global__ void k_zero_c(float* ws) {
  ws[OFS_C + blockIdx.x * 256 + threadIdx.x] = 0.f;
}

__global__ void k_invc(float* ws) {
  int idx = blockIdx.x * 256 + threadIdx.x;
  ws[OFS_C + idx] = 1.0f / ws[OFS_C + idx];
}

// ---------------------------------------------------------------- K1: gl, sa, sb
__global__ void k_nodelin(const float* x, const float* W1, const float* b1, float* ws) {
  __shared__ float xs[8][F_];
  int t = threadIdx.x;
  int r = t >> 5, lane = t & 31;
  int row = blockIdx.x * 8 + r;            // row in [0, B*N)
  xs[r][lane] = x[row * F_ + lane];
  __syncthreads();
  float acc = b1[lane];
  for (int f = 0; f < F_; ++f) acc += xs[r][f] * W1[f * H_ + lane];
  ws[OFS_GL + row * H_ + lane] = acc;
  if (lane < P_) {
    float s = 0.f;
    for (int f = 0; f < F_; ++f) s += xs[r][f] * ws[OFS_V2 + lane * F_ + f];
    ws[OFS_SA + lane * (B_ * N_) + row] = s;
  } else if (lane < 2 * P_) {
    int p = lane - P_;
    float s = 0.f;
    for (int f = 0; f < F_; ++f) s += xs[r][f] * ws[OFS_V3 + p * F_ + f];
    ws[OFS_SB + p * (B_ * N_) + row] = s;
  }
}

// ---------------------------------------------------------------- K2: row sums -> 1/r
__global__ void k_rowsum(const float* e, float* ws) {
  int b = blockIdx.x >> 10, i = blockIdx.x & 1023;
  int t = threadIdx.x;
  float bp[P_];
  for (int p = 0; p < P_; ++p)
    bp[p] = ws[OFS_BP + p] + ws[OFS_SB + p * (B_ * N_) + b * N_ + i];
  float acc[P_] = {0.f, 0.f, 0.f, 0.f};
  const float4* ep = (const float4*)(e + ((size_t)(b * N_ + i)) * N_ * P_);
  for (int j = t; j < N_; j += 256) {
    float4 ev = ep[j];
    float evv[4] = {ev.x, ev.y, ev.z, ev.w};
    for (int p = 0; p < P_; ++p) {
      float coef = fmaxf(ws[OFS_SA + p * (B_ * N_) + b * N_ + j] + bp[p], 0.f);
      acc[p] += coef * evv[p];
    }
  }
  __shared__ float red[P_][256];
  for (int p = 0; p < P_; ++p) red[p][t] = acc[p];
  __syncthreads();
  for (int s = 128; s > 0; s >>= 1) {
    if (t < s) for (int p = 0; p < P_; ++p) red[p][t] += red[p][t + s];
    __syncthreads();
  }
  if (t < P_) ws[OFS_R + (b * P_ + t) * N_ + i] = 1.0f / red[t][0];
}

// ---------------------------------------------------------------- K3a: column sums of m/r
__global__ void k_colsum(const float* e, float* ws) {
  int blk = blockIdx.x;           // b(8) x jc(4) x ic(8)
  int b = blk >> 5;
  int jc = (blk >> 3) & 3;
  int ic = blk & 7;
  int t = threadIdx.x;
  int j = jc * 256 + t;
  int i0 = ic * 128;
  __shared__ float sbs[P_][128], rin[P_][128];
  for (int u = t; u < P_ * 128; u += 256) {
    int p = u >> 7, ii = u & 127;
    sbs[p][ii] = ws[OFS_SB + p * (B_ * N_) + b * N_ + i0 + ii];
    rin[p][ii] = ws[OFS_R + (b * P_ + p) * N_ + i0 + ii];
  }
  __syncthreads();
  float sav[P_];
  for (int p = 0; p < P_; ++p)
    sav[p] = ws[OFS_SA + p * (B_ * N_) + b * N_ + j] + ws[OFS_BP + p];
  float cs[P_] = {0.f, 0.f, 0.f, 0.f};
  for (int ii = 0; ii < 128; ++ii) {
    float4 ev = *(const float4*)(e + (((size_t)(b * N_ + i0 + ii)) * N_ + j) * P_);
    float evv[4] = {ev.x, ev.y, ev.z, ev.w};
    for (int p = 0; p < P_; ++p) {
      float coef = fmaxf(sav[p] + sbs[p][ii], 0.f);
      cs[p] += coef * evv[p] * rin[p][ii];
    }
  }
  for (int p = 0; p < P_; ++p)
    atomicAdd(&ws[OFS_C + (b * P_ + p) * N_ + j], cs[p]);
}

// ---------------------------------------------------------------- K3b: pack S = m/(r_i c_j)
__global__ void k_scale(const float* e, float* ws) {
  int b = blockIdx.x >> 10, i = blockIdx.x & 1023;
  int t = threadIdx.x;
  float sbv[P_], rin[P_];
  for (int p = 0; p < P_; ++p) {
    sbv[p] = ws[OFS_SB + p * (B_ * N_) + b * N_ + i] + ws[OFS_BP + p];
    rin[p] = ws[OFS_R + (b * P_ + p) * N_ + i];
  }
  const float4* ep = (const float4*)(e + ((size_t)(b * N_ + i)) * N_ * P_);
  for (int j = t; j < N_; j += 256) {
    float4 ev = ep[j];
    float evv[4] = {ev.x, ev.y, ev.z, ev.w};
    for (int p = 0; p < P_; ++p) {
      float coef = fmaxf(ws[OFS_SA + p * (B_ * N_) + b * N_ + j] + sbv[p], 0.f);
      float s = coef * evv[p] * rin[p] * ws[OFS_C + (b * P_ + p) * N_ + j];
      ws[OFS_S + ((size_t)(b * P_ + p) * N_ + i) * N_ + j] = s;
    }
  }
}

// ---------------------------------------------------------------- K4: new_e = S @ S^T (fp32 WMMA, symmetric)
#define KT    16
#define PITCH 20                       // dword pitch; 16B-aligned rows, bank-conflict-free b64 frags
#define SLABF (P_ * 64 * PITCH)        // 5120 floats per slab

__global__ __launch_bounds__(256) void k_bmm(const float* ws_, float* out) {
  const float* S = ws_ + OFS_S;
  float* newE = out + RESX_N;
  __shared__ __align__(16) float lds[2 * SLABF];   // 40 KB; reused for output staging

  int blk = blockIdx.x;
  int b = blk / 136;
  int pr = blk % 136;                 // upper-triangular pair index over 16x16 macro tiles
  int ti = 0, rem = pr;
  while (rem >= (16 - ti)) { rem -= 16 - ti; ++ti; }
  int tk = ti + rem;
  int i0 = ti * 64, k0 = tk * 64;

  int t = threadIdx.x;
  int wave = t >> 5, lane = t & 31;
  int hs = lane >> 4, lr = lane & 15;
  int is_ = wave >> 1, ks0 = (wave & 1) * 2;

  v8f acc[2][P_];
  for (int q = 0; q < 2; ++q) for (int p = 0; p < P_; ++p) acc[q][p] = v8f_zero();

  size_t baseA = (size_t)b * P_ * N_ * (size_t)N_;
  for (int kb = 0; kb < N_; kb += KT) {
    for (int u = t; u < 2048; u += 256) {        // 2 slabs * 4p * 64 rows * 4 float4s
      int c4 = u & 3;
      int r = (u >> 2) & 63;
      int p = (u >> 8) & 3;
      int sel = u >> 10;
      int rowg = (sel ? k0 : i0) + r;
      float4 v = *(const float4*)(S + baseA + ((size_t)p * N_ + rowg) * N_ + kb + c4 * 4);
      *(float4*)(lds + sel * SLABF + (p * 64 + r) * PITCH + c4 * 4) = v;
    }
    __syncthreads();
    for (int jj = 0; jj < KT; jj += 4) {
      int cofs = jj + 2 * hs;
      v2f a[P_];
      for (int p = 0; p < P_; ++p)
        a[p] = *(const v2f*)(lds + (p * 64 + is_ * 16 + lr) * PITCH + cofs);
      for (int q = 0; q < 2; ++q) {
        for (int p = 0; p < P_; ++p) {
          v2f bf = *(const v2f*)(lds + SLABF + (p * 64 + (ks0 + q) * 16 + lr) * PITCH + cofs);
          acc[q][p] = __builtin_amdgcn_wmma_f32_16x16x4_f32(
              false, a[p], false, bf, (short)0, acc[q][p], false, false);
        }
      }
    }
    __syncthreads();
  }

  // epilogue: stage 32 i-rows at a time as [il][kl][p] float4, write both orientations coalesced
  for (int half = 0; half < 2; ++half) {
    if ((is_ >> 1) == half) {
      for (int q = 0; q < 2; ++q) {
        int kl = (ks0 + q) * 16 + lr;
        for (int v = 0; v < 8; ++v) {
          int il = is_ * 16 + v + 8 * hs - half * 32;
          float4 w = make_float4(acc[q][0][v], acc[q][1][v], acc[q][2][v], acc[q][3][v]);
          *(float4*)(lds + (il * 64 + kl) * 4) = w;
        }
      }
    }
    __syncthreads();
    for (int u = t; u < 2048; u += 256) {
      int il = u >> 6, kl = u & 63;
      float4 w = *(const float4*)(lds + u * 4);
      *(float4*)(newE + (((size_t)(b * N_ + i0 + half * 32 + il)) * N_ + k0 + kl) * P_) = w;
    }
    if (ti != tk) {
      for (int u = t; u < 2048; u += 256) {
        int kl = u >> 5, il = u & 31;
        float4 w = *(const float4*)(lds + (il * 64 + kl) * 4);
        *(float4*)(newE + (((size_t)(b * N_ + k0 + kl)) * N_ + i0 + half * 32 + il) * P_) = w;
      }
    }
    __syncthreads();
  }
}

// ---------------------------------------------------------------- K5: res_x = (coef @ gl) @ Wlast
__global__ void k_out(const float* ws_, const float* Wlast, const float* blast, float* out) {
  const float* gl = ws_ + OFS_GL;
  int b = blockIdx.x >> 6;
  int i0 = (blockIdx.x & 63) * 16;
  int lane = threadIdx.x;
  int hs = lane >> 4, lr = lane & 15;
  __shared__ float xt[16 * 130];      // 16 x 128 staged xagg tile, padded pitch 130

  float sbl[P_], bp;
  for (int p = 0; p < P_; ++p)
    sbl[p] = ws_[OFS_SB + p * (B_ * N_) + b * N_ + i0 + lr] + ws_[OFS_BP + p];
  (void)bp;

  v8f acc[P_][2];
  for (int p = 0; p < P_; ++p) for (int h = 0; h < 2; ++h) acc[p][h] = v8f_zero();

  for (int j = 0; j < N_; j += 4) {
    int c0 = j + 2 * hs;
    v2f af[P_];
    for (int p = 0; p < P_; ++p) {
      v2f sj = *(const v2f*)(ws_ + OFS_SA + p * (B_ * N_) + b * N_ + c0);
      af[p].x = fmaxf(sj.x + sbl[p], 0.f);
      af[p].y = fmaxf(sj.y + sbl[p], 0.f);
    }
    for (int hsub = 0; hsub < 2; ++hsub) {
      v2f bf;
      bf.x = gl[(b * N_ + c0) * H_ + hsub * 16 + lr];
      bf.y = gl[(b * N_ + c0 + 1) * H_ + hsub * 16 + lr];
      for (int p = 0; p < P_; ++p)
        acc[p][hsub] = __builtin_amdgcn_wmma_f32_16x16x4_f32(
            false, af[p], false, bf, (short)0, acc[p][hsub], false, false);
    }
  }
  for (int p = 0; p < P_; ++p)
    for (int hsub = 0; hsub < 2; ++hsub)
      for (int v = 0; v < 8; ++v) {
        int M = v + 8 * hs;
        xt[M * 130 + p * 32 + hsub * 16 + lr] = acc[p][hsub][v];
      }
  __syncthreads();

  v8f oacc[4];
  for (int os = 0; os < 4; ++os) oacc[os] = v8f_zero();
  for (int kk = 0; kk < P_ * H_; kk += 4) {
    int c0 = kk + 2 * hs;
    v2f af = *(const v2f*)(xt + lr * 130 + c0);
    for (int os = 0; os < 4; ++os) {
      v2f bf;
      bf.x = Wlast[c0 * OUT_ + os * 16 + lr];
      bf.y = Wlast[(c0 + 1) * OUT_ + os * 16 + lr];
      oacc[os] = __builtin_amdgcn_wmma_f32_16x16x4_f32(
          false, af, false, bf, (short)0, oacc[os], false, false);
    }
  }
  for (int os = 0; os < 4; ++os)
    for (int v = 0; v < 8; ++v) {
      int row = i0 + v + 8 * hs;
      int col = os * 16 + lr;
      out[((size_t)b * N_ + row) * OUT_ + col] = oacc[os][v] + blast[col];
    }
}

// ---------------------------------------------------------------- launch
extern "C" void kernel_launch(void* const* d_in, const int* in_sizes, int n_in,
                              void* d_out, int out_size, void* d_ws, size_t ws_size,
                              hipStream_t stream) {
  (void)in_sizes; (void)n_in; (void)out_size; (void)ws_size;
  const float* x     = (const float*)d_in[0];
  const float* e     = (const float*)d_in[1];
  const float* W1    = (const float*)d_in[2];
  const float* b1    = (const float*)d_in[3];
  const float* W2    = (const float*)d_in[4];
  const float* b2    = (const float*)d_in[5];
  const float* W3    = (const float*)d_in[6];
  const float* b3    = (const float*)d_in[7];
  const float* w4a   = (const float*)d_in[8];
  const float* w4b   = (const float*)d_in[9];
  const float* b4    = (const float*)d_in[10];
  const float* Wlast = (const float*)d_in[11];
  const float* blast = (const float*)d_in[12];
  float* out = (float*)d_out;
  float* ws  = (float*)d_ws;

  k_prep   <<<1, 256, 0, stream>>>(W2, b2, W3, b3, w4a, w4b, b4, ws);
  k_zero_c <<<128, 256, 0, stream>>>(ws);
  k_nodelin<<<B_ * N_ / 8, 256, 0, stream>>>(x, W1, b1, ws);
  k_rowsum <<<B_ * N_, 256, 0, stream>>>(e, ws);
  k_colsum <<<256, 256, 0, stream>>>(e, ws);
  k_invc   <<<128, 256, 0, stream>>>(ws);
  k_scale  <<<B_ * N_, 256, 0, stream>>>(e, ws);
  k_bmm    <<<B_ * 136, 256, 0, stream>>>(ws, out);
  k_out    <<<B_ * (N_ / 16), 32, 0, stream>>>(ws, Wlast, blast, out);
}